// IncrSeqAttnMatch_16209206575369
// MI455X (gfx1250) — compile-verified
//
#include <hip/hip_runtime.h>
#include <hip/hip_bf16.h>

typedef __attribute__((ext_vector_type(2))) float v2f;
typedef __attribute__((ext_vector_type(8))) float v8f;

constexpr int Tn = 64;    // turns
constexpr int LQ = 64;    // question length
constexpr int LA = 64;    // answer length
constexpr int Hd = 512;   // input size
constexpr int Kd = 256;   // hidden (projection) size
constexpr int Bb = 128;   // tokens per turn (LQ+LA)
constexpr int Sl = 8192;  // full history length Tn*Bb
constexpr int PSTR = 132; // LDS row stride (floats): conflict-free WMMA A reads
constexpr int NBMAX = 32; // max split-K workgroups per turn

// -------------------------------------------------------------------------
// Kernel 1: keys = relu(X @ W^T + b), X = interleaved [xq;xa] per turn.
// One 16x16 output tile per wave, K-loop of 128 chained f32 WMMAs (K=4).
// -------------------------------------------------------------------------
__global__ __launch_bounds__(256) void proj_kernel(
    const float* __restrict__ xq, const float* __restrict__ xa,
    const float* __restrict__ W, const float* __restrict__ bvec,
    float* __restrict__ keys) {
  const int lane = threadIdx.x & 31;
  const int wave = threadIdx.x >> 5;
  const int tile = blockIdx.x * 8 + wave;      // 8192 tiles = (S/16)*(K/16)
  const int srow = (tile >> 4) << 4;
  const int kcol = (tile & 15) << 4;
  const int half = lane >> 4;
  const int l16  = lane & 15;
  const int d    = half * 2;

  const int s   = srow + l16;
  const int t   = s >> 7;
  const int idx = s & 127;
  const float* arow = (idx < LQ) ? (xq + (size_t)(t * LQ + idx) * Hd)
                                 : (xa + (size_t)(t * LA + (idx - LQ)) * Hd);
  const float* brow = W + (size_t)(kcol + l16) * Hd;

  v8f c = {0.f, 0.f, 0.f, 0.f, 0.f, 0.f, 0.f, 0.f};
  for (int h0 = 0; h0 < Hd; h0 += 4) {
    v2f a  = *(const v2f*)(arow + h0 + d);
    v2f bw = *(const v2f*)(brow + h0 + d);
    c = __builtin_amdgcn_wmma_f32_16x16x4_f32(false, a, false, bw,
                                              (short)0, c, false, false);
  }
  const float bias = bvec[kcol + l16];
#pragma unroll
  for (int i = 0; i < 8; ++i) {
    float v = c[i] + bias;
    v = v > 0.f ? v : 0.f;
    const int M = half * 8 + i;
    keys[(size_t)(srow + M) * Kd + (kcol + l16)] = v;
  }
}

// -------------------------------------------------------------------------
// Kernel 2: seed values[0:128] with turn-0 block, out[0] = xq[0],
//           kbias[s] = pad ? -big : 0 (softmax mask as additive bias).
// -------------------------------------------------------------------------
__global__ void init_kernel(const float* __restrict__ xq, const float* __restrict__ xa,
                            const unsigned char* __restrict__ qm,
                            const unsigned char* __restrict__ am,
                            float* __restrict__ out, float* __restrict__ values,
                            float* __restrict__ kbias) {
  const int i = blockIdx.x * blockDim.x + threadIdx.x;   // 65536 threads
  if (i < Bb * Hd) {
    const int row = i / Hd, col = i % Hd;
    const float v = (row < LQ) ? xq[row * Hd + col] : xa[(row - LQ) * Hd + col];
    values[i] = v;
    if (row < LQ) out[i] = v;
  }
  if (i < Sl) {
    const int t = i >> 7, idx = i & 127;
    const unsigned char p = (idx < LQ) ? qm[t * LQ + idx] : am[t * LA + (idx - LQ)];
    kbias[i] = p ? -3.0e38f : 0.0f;
  }
}

// -------------------------------------------------------------------------
// Kernel 3: per-turn split-K flash attention partials. Block b handles key
// chunks c = b, b+nb, ... (each chunk = one 128-key turn block). Emits
// per-block (m, l, unnormalized acc) for the combine kernel.
// -------------------------------------------------------------------------
__global__ __launch_bounds__(512) void attn_part_kernel(
    const float* __restrict__ keys, const float* __restrict__ kbias,
    const float* __restrict__ values, int t, int nb,
    float* __restrict__ pm, float* __restrict__ pl, float* __restrict__ pacc) {
  __shared__ float P[64 * PSTR];
  __shared__ float m_sh[64], l_sh[64], r_sh[64];

  const int tid  = threadIdx.x;
  const int lane = tid & 31, wave = tid >> 5;
  const int half = lane >> 4, l16 = lane & 15;
  const int d    = half * 2;
  const int bid  = blockIdx.x;

  const int qt_pv = wave & 3;          // PV: wave -> (query tile, 8 h-tiles)
  const int hb    = (wave >> 2) * 8;

  if (tid < 64) { m_sh[tid] = -3.0e38f; l_sh[tid] = 0.f; }
  v8f acc[8];
#pragma unroll
  for (int i = 0; i < 8; ++i) acc[i] = (v8f){0.f,0.f,0.f,0.f,0.f,0.f,0.f,0.f};
  __syncthreads();

  const float* qbase = keys + (size_t)t * Bb * Kd;   // q_proj rows of this turn

  for (int c = bid; c < t; c += nb) {
    const int ks0 = c * Bb;

    // ---- phase 1: S = Q.K^T (+mask bias), two 16x16 tiles per wave ----
#pragma unroll
    for (int sub = 0; sub < 2; ++sub) {
      const int id = wave * 2 + sub;                  // 4 qtiles x 8 ktiles
      const int qt = id >> 3, kt = id & 7;
      const float* arow = qbase + (size_t)(qt * 16 + l16) * Kd;
      const float* brow = keys + (size_t)(ks0 + kt * 16 + l16) * Kd;
      v8f sc = {0.f,0.f,0.f,0.f,0.f,0.f,0.f,0.f};
      for (int k0 = 0; k0 < Kd; k0 += 4) {
        v2f a  = *(const v2f*)(arow + k0 + d);
        v2f bk = *(const v2f*)(brow + k0 + d);
        sc = __builtin_amdgcn_wmma_f32_16x16x4_f32(false, a, false, bk,
                                                   (short)0, sc, false, false);
      }
      const float kb = kbias[ks0 + kt * 16 + l16];
#pragma unroll
      for (int i = 0; i < 8; ++i) {
        const int M = qt * 16 + half * 8 + i;
        P[M * PSTR + kt * 16 + l16] = sc[i] + kb;
      }
    }
    __syncthreads();

    // ---- phase 2: online softmax update, one thread per query row ----
    if (tid < 64) {
      float* row = &P[tid * PSTR];
      const float m_old = m_sh[tid];
      float cm = m_old;
      for (int j = 0; j < Bb; ++j) cm = fmaxf(cm, row[j]);
      const float r = __expf(m_old - cm);
      float sum = 0.f;
      for (int j = 0; j < Bb; ++j) {
        const float p = __expf(row[j] - cm);
        row[j] = p;
        sum += p;
      }
      m_sh[tid] = cm;
      l_sh[tid] = l_sh[tid] * r + sum;
      r_sh[tid] = r;
    }
    __syncthreads();

    // ---- phase 3: acc = acc*r + P.V (A from LDS, B from values) ----
    float rr[8];
#pragma unroll
    for (int i = 0; i < 8; ++i) rr[i] = r_sh[qt_pv * 16 + half * 8 + i];
#pragma unroll
    for (int hh = 0; hh < 8; ++hh)
#pragma unroll
      for (int i = 0; i < 8; ++i) acc[hh][i] *= rr[i];

    const float* vb = values + (size_t)ks0 * Hd;
    for (int kk = 0; kk < Bb; kk += 4) {
      const v2f a = *(const v2f*)(&P[(qt_pv * 16 + l16) * PSTR + kk + d]);
#pragma unroll
      for (int hh = 0; hh < 8; ++hh) {
        const int h0 = (hb + hh) * 16;
        const float* v0 = vb + (size_t)(kk + d) * Hd + h0 + l16;
        v2f bv;
        bv[0] = v0[0];
        bv[1] = v0[Hd];
        acc[hh] = __builtin_amdgcn_wmma_f32_16x16x4_f32(false, a, false, bv,
                                                        (short)0, acc[hh], false, false);
      }
    }
    __syncthreads();      // protect P before next chunk overwrites it
  }

  // ---- emit partials ----
  if (tid < 64) {
    pm[bid * 64 + tid] = m_sh[tid];
    pl[bid * 64 + tid] = l_sh[tid];
  }
#pragma unroll
  for (int hh = 0; hh < 8; ++hh) {
    const int h0 = (hb + hh) * 16 + l16;
#pragma unroll
    for (int i = 0; i < 8; ++i) {
      const int q = qt_pv * 16 + half * 8 + i;
      pacc[((size_t)bid * LQ + q) * Hd + h0] = acc[hh][i];
    }
  }
}

// -------------------------------------------------------------------------
// Kernel 4: merge split-K partials, finalize xq_plus, append value block.
// -------------------------------------------------------------------------
__global__ __launch_bounds__(512) void combine_kernel(
    const float* __restrict__ xq, const float* __restrict__ xa,
    const float* __restrict__ pm, const float* __restrict__ pl,
    const float* __restrict__ pacc, int t, int nb,
    float* __restrict__ values, float* __restrict__ out) {
  __shared__ float msh[64];
  __shared__ float minv[64];            // 1/l per row
  __shared__ float fw[NBMAX * 64];      // exp(m_b - m) per (block,row)

  const int tid = threadIdx.x;
  if (tid < 64) {
    float m = -3.0e38f;
    for (int b = 0; b < nb; ++b) m = fmaxf(m, pm[b * 64 + tid]);
    msh[tid] = m;
  }
  __syncthreads();
  for (int j = tid; j < nb * 64; j += 512) {
    const int q = j & 63;
    fw[j] = __expf(pm[j] - msh[q]);
  }
  __syncthreads();
  if (tid < 64) {
    float l = 0.f;
    for (int b = 0; b < nb; ++b) l += pl[b * 64 + tid] * fw[b * 64 + tid];
    minv[tid] = 1.0f / l;
  }
  __syncthreads();

  const float* xqt = xq + (size_t)t * LQ * Hd;
  float* outt = out + (size_t)t * LQ * Hd;
  float* valt = values + (size_t)t * Bb * Hd;
  for (int i = tid; i < LQ * Hd; i += 512) {
    const int q = i >> 9, h = i & (Hd - 1);
    float s = 0.f;
    for (int b = 0; b < nb; ++b)
      s += pacc[((size_t)b * LQ + q) * Hd + h] * fw[b * 64 + q];
    const float xp = 0.5f * (xqt[i] + s * minv[q]);
    outt[i] = xp;
    valt[i] = xp;
  }
  // answers appended unmodified
  const float* xat = xa + (size_t)t * LA * Hd;
  float* vala = valt + (size_t)LQ * Hd;
  for (int i = tid; i < LA * Hd; i += 512) vala[i] = xat[i];
}

extern "C" void kernel_launch(void* const* d_in, const int* in_sizes, int n_in,
                              void* d_out, int out_size, void* d_ws, size_t ws_size,
                              hipStream_t stream) {
  (void)in_sizes; (void)n_in; (void)out_size; (void)ws_size;
  const float* xq = (const float*)d_in[0];
  const float* xa = (const float*)d_in[1];
  const unsigned char* qm = (const unsigned char*)d_in[2];
  const unsigned char* am = (const unsigned char*)d_in[3];
  const float* W  = (const float*)d_in[4];
  const float* b  = (const float*)d_in[5];
  float* out = (float*)d_out;

  // Workspace: keys [S,K] | values [S,H] | kbias [S] | pm | pl | pacc  (~28 MB)
  float* keys   = (float*)d_ws;
  float* values = keys + (size_t)Sl * Kd;
  float* kbias  = values + (size_t)Sl * Hd;
  float* pm     = kbias + Sl;
  float* pl     = pm + NBMAX * 64;
  float* pacc   = pl + NBMAX * 64;

  hipLaunchKernelGGL(proj_kernel, dim3(1024), dim3(256), 0, stream, xq, xa, W, b, keys);
  hipLaunchKernelGGL(init_kernel, dim3(256), dim3(256), 0, stream, xq, xa, qm, am, out, values, kbias);

  for (int t = 1; t < Tn; ++t) {
    const int nb = (t < NBMAX) ? t : NBMAX;
    hipLaunchKernelGGL(attn_part_kernel, dim3(nb), dim3(512), 0, stream,
                       keys, kbias, values, t, nb, pm, pl, pacc);
    hipLaunchKernelGGL(combine_kernel, dim3(1), dim3(512), 0, stream,
                       xq, xa, pm, pl, pacc, t, nb, values, out);
  }
}